// RankLoss_59390807769109
// MI455X (gfx1250) — compile-verified
//
#include <hip/hip_runtime.h>

typedef __attribute__((ext_vector_type(2))) float v2f;
typedef __attribute__((ext_vector_type(8))) float v8f;
typedef __attribute__((ext_vector_type(4))) int   i32x4;

#define BDIM 256
#define CDIM 1024
#define ROWS_PB 16
#define WAVES 8
#define CHUNK 256                 // columns staged per iteration
#define NCHUNK (CDIM / CHUNK)
#define LDS_STRIDE 260            // floats per row: 256 + 4 pad (16B) -> bank-conflict-free reads

typedef __attribute__((address_space(1))) i32x4 gvec_t;   // global int4
typedef __attribute__((address_space(3))) i32x4 lvec_t;   // LDS int4

// Coalesced 16B global -> LDS async copy (ASYNCcnt path, gfx1250).
__device__ __forceinline__ void async_b128(const void* g, void* l) {
    __builtin_amdgcn_global_load_async_to_lds_b128((gvec_t*)g, (lvec_t*)l, 0, 0);
}

// Kernel 1: per-row sums via V_WMMA_F32_16X16X4_F32 (exact f32 accumulate).
// Block = 16 rows; 8 waves split each staged 256-col chunk (32 cols/wave).
// A tile (16x4 f32): lane L<16 -> (row L, k+0..1), lane L>=16 -> (row L-16, k+2..3).
// B (4x16 f32): ones in column N=0 -> lanes 0 and 16 hold 1.0 in both VGPRs.
// D(m,0) lands in acc[m&7] of lane 0 (m<8) / lane 16 (m>=8).
__global__ __launch_bounds__(256) void rank_row_stats(const float* __restrict__ inp,
                                                      const int*   __restrict__ tgt,
                                                      float* __restrict__ ws_loss,
                                                      float* __restrict__ ws_valid)
{
    __shared__ float ldsIn[ROWS_PB][LDS_STRIDE];   // 16.6 KB
    __shared__ int   ldsTg[ROWS_PB][LDS_STRIDE];   // 16.6 KB
    __shared__ float part[WAVES][3][32][8];        // 24 KB partial accumulators

    const int tid   = threadIdx.x;
    const int wave  = tid >> 5;
    const int lane  = tid & 31;
    const int row16 = lane & 15;
    const int khalf = (lane >> 4) << 1;            // 0 or 2
    const int rowBase = blockIdx.x * ROWS_PB;

    v8f acc_s  = {0.f,0.f,0.f,0.f,0.f,0.f,0.f,0.f};
    v8f acc_st = acc_s;
    v8f acc_t  = acc_s;

    v2f bcol;
    bcol.x = (lane == 0 || lane == 16) ? 1.0f : 0.0f;   // B[k][0] = 1 for all k
    bcol.y = bcol.x;

    for (int ch = 0; ch < NCHUNK; ++ch) {
        const int colBase = ch * CHUNK;

        // Stage 16 rows x 256 cols of input + target, fully coalesced:
        // 1024 16B-chunks per tile, 256 threads -> 4 rounds each.
        #pragma unroll
        for (int r = 0; r < 4; ++r) {
            const int u  = r * BDIM + tid;     // 16B-chunk id within tile
            const int rw = u >> 6;             // row (64 x 16B per row)
            const int c4 = (u & 63) << 2;      // column in floats
            async_b128(inp + (size_t)(rowBase + rw) * CDIM + colBase + c4, &ldsIn[rw][c4]);
            async_b128(tgt + (size_t)(rowBase + rw) * CDIM + colBase + c4, &ldsTg[rw][c4]);
        }
        asm volatile("s_wait_asynccnt 0x0" ::: "memory");
        __syncthreads();

        const int kw0 = wave * (CHUNK / WAVES);
        #pragma unroll
        for (int k = 0; k < CHUNK / WAVES; k += 4) {
            const int c = kw0 + k + khalf;
            const float x0 = ldsIn[row16][c];
            const float x1 = ldsIn[row16][c + 1];
            const int   t0 = ldsTg[row16][c];
            const int   t1 = ldsTg[row16][c + 1];

            // sigmoid via v_exp_f32 + v_rcp_f32 (fast path; ~1 ulp)
            const float s0 = __builtin_amdgcn_rcpf(1.0f + __expf(-x0));
            const float s1 = __builtin_amdgcn_rcpf(1.0f + __expf(-x1));

            v2f a_s  = {s0, s1};
            v2f a_st = {(t0 != 0) ? s0 : 0.0f, (t1 != 0) ? s1 : 0.0f};
            v2f a_t  = {(float)(t0 != 0), (float)(t1 != 0)};

            // D = A x B + C, 8 args: (neg_a, A, neg_b, B, c_mod, C, reuse_a, reuse_b)
            acc_s  = __builtin_amdgcn_wmma_f32_16x16x4_f32(false, a_s,  false, bcol, (short)0, acc_s,  false, false);
            acc_st = __builtin_amdgcn_wmma_f32_16x16x4_f32(false, a_st, false, bcol, (short)0, acc_st, false, false);
            acc_t  = __builtin_amdgcn_wmma_f32_16x16x4_f32(false, a_t,  false, bcol, (short)0, acc_t,  false, false);
        }
        __syncthreads();   // before next chunk overwrites the staging tiles
    }

    #pragma unroll
    for (int j = 0; j < 8; ++j) {
        part[wave][0][lane][j] = acc_s[j];
        part[wave][1][lane][j] = acc_st[j];
        part[wave][2][lane][j] = acc_t[j];
    }
    __syncthreads();

    if (tid < ROWS_PB) {
        const int srcLane = (tid < 8) ? 0 : 16;   // where D(row,0) lives
        const int idx     = tid & 7;
        float sum_s = 0.f, sum_st = 0.f, sum_t = 0.f;
        #pragma unroll
        for (int w = 0; w < WAVES; ++w) {
            sum_s  += part[w][0][srcLane][idx];
            sum_st += part[w][1][srcLane][idx];
            sum_t  += part[w][2][srcLane][idx];
        }
        // Sum of 0/1 over 1024 in f32 WMMA accumulate is exact.
        const int  npos  = (int)(sum_t + 0.5f);
        const bool valid = (npos > 0) && (npos < CDIM);

        float rowloss = 0.0f;
        if (valid) {
            // cond = (s[n]-s[p]+1 >= 0) always holds since sigmoid in [0,1]:
            // loss_i = sum_neg + nneg - nneg*sum_pos/npos
            const float nneg    = (float)(CDIM - npos);
            const float sum_neg = sum_s - sum_st;
            rowloss = sum_neg + nneg - nneg * sum_st / (float)npos;
        }
        ws_loss[rowBase + tid]  = rowloss;
        ws_valid[rowBase + tid] = valid ? 1.0f : 0.0f;
    }
}

// Kernel 2: deterministic tree reduction of 256 row losses + valid count.
__global__ __launch_bounds__(256) void rank_finalize(const float* __restrict__ ws_loss,
                                                     const float* __restrict__ ws_valid,
                                                     float* __restrict__ out)
{
    __shared__ float sl[256];
    __shared__ float sv[256];
    const int t = threadIdx.x;
    sl[t] = ws_loss[t];
    sv[t] = ws_valid[t];
    __syncthreads();
    for (int off = 128; off > 0; off >>= 1) {
        if (t < off) {
            sl[t] += sl[t + off];
            sv[t] += sv[t + off];
        }
        __syncthreads();
    }
    if (t == 0) {
        const float bn = (sv[0] == 0.0f) ? 1.0f : sv[0];
        out[0] = sl[0] / bn;
    }
}

extern "C" void kernel_launch(void* const* d_in, const int* in_sizes, int n_in,
                              void* d_out, int out_size, void* d_ws, size_t ws_size,
                              hipStream_t stream)
{
    const float* inp = (const float*)d_in[0];   // [256, 1024] f32
    const int*   tgt = (const int*)d_in[1];     // [256, 1024] i32
    float* out = (float*)d_out;                 // [1] f32

    float* ws_loss  = (float*)d_ws;             // 256 floats
    float* ws_valid = ws_loss + BDIM;           // 256 floats

    rank_row_stats<<<BDIM / ROWS_PB, 256, 0, stream>>>(inp, tgt, ws_loss, ws_valid);
    rank_finalize<<<1, 256, 0, stream>>>(ws_loss, ws_valid, out);
}